// GraphAE_82042465288476
// MI455X (gfx1250) — compile-verified
//
#include <hip/hip_runtime.h>

typedef __attribute__((ext_vector_type(2))) float v2f;
typedef __attribute__((ext_vector_type(8))) float v8f;

#define N_TXN_C 100000
#define N_CLI_C 20000
#define D_TXN_C 64
#define D_CLI_C 32
#define H_C     128

// Native no-return f32 global atomic add (global_atomic_add_f32, STOREcnt
// tracked) — relaxed ordering, device scope; avoids any CAS-loop fallback.
__device__ __forceinline__ void atomic_add_f32(float* p, float v) {
  (void)__hip_atomic_fetch_add(p, v, __ATOMIC_RELAXED, __HIP_MEMORY_SCOPE_AGENT);
}

// ---------------------------------------------------------------------------
// Zero-fill (re-initializes scatter accumulators every call; harness poisons
// ws once and never re-poisons, so this must run at the top of each launch).
// ---------------------------------------------------------------------------
__global__ void zero_f32(float* __restrict__ p, long long n) {
  long long i = (long long)blockIdx.x * blockDim.x + threadIdx.x;
  long long stride = (long long)gridDim.x * blockDim.x;
  for (; i < n; i += stride) p[i] = 0.0f;
}

// ---------------------------------------------------------------------------
// Edge scatter: one wave (32 lanes) per edge. Vectorized coalesced gather of
// the source row (b128/b64/b32 per lane depending on D), then native f32
// global atomics into the destination accumulator. Optionally bumps the
// destination in-degree count (lane 0).
// ---------------------------------------------------------------------------
__global__ void scatter_add(const float* __restrict__ xsrc,
                            const int* __restrict__ src,
                            const int* __restrict__ dst,
                            float* __restrict__ agg,
                            float* __restrict__ cnt,
                            int E, int D) {
  int t = blockIdx.x * blockDim.x + threadIdx.x;
  int e = t >> 5;
  int lane = t & 31;
  if (e >= E) return;
  int s = src[e];
  int d = dst[e];
  const float* xp = xsrc + (long long)s * D;
  float* ap = agg + (long long)d * D;

  if (D == 128) {            // lane covers 4 contiguous floats: one b128 load
    float4 v = ((const float4*)xp)[lane];
    float* a = ap + lane * 4;
    atomic_add_f32(a + 0, v.x);
    atomic_add_f32(a + 1, v.y);
    atomic_add_f32(a + 2, v.z);
    atomic_add_f32(a + 3, v.w);
  } else if (D == 64) {      // lane covers 2 floats: one b64 load
    float2 v = ((const float2*)xp)[lane];
    float* a = ap + lane * 2;
    atomic_add_f32(a + 0, v.x);
    atomic_add_f32(a + 1, v.y);
  } else {                   // D == 32: one float per lane
    atomic_add_f32(ap + lane, xp[lane]);
  }

  if (cnt != nullptr && lane == 0) {
    atomic_add_f32(cnt + d, 1.0f);
  }
}

// ---------------------------------------------------------------------------
// fp32 WMMA: D = A(16x4) * B(4x16) + C, exact fp32 (matches fp32 reference).
// ---------------------------------------------------------------------------
__device__ __forceinline__ v8f wmma_f32_k4(v2f a, v2f b, v8f c) {
  // 8 args: (neg_a, A, neg_b, B, c_mod, C, reuse_a, reuse_b)
  return __builtin_amdgcn_wmma_f32_16x16x4_f32(
      false, a, false, b, (short)0, c, false, false);
}

// ---------------------------------------------------------------------------
// Fused SAGE / dense layer:
//   out[m, n] = act( (A1[m,:] * inv_cnt[m]) @ W1 + (A2 ? A2[m,:] @ W2 : 0) + bias[n] )
// Block = 128 threads (4 waves), covers 16 output rows. Each wave owns 16x16
// output tiles (N/16 tiles round-robin over 4 waves). A rows staged in LDS
// with the mean normalization applied; W fragments read straight from global
// (weights are tiny and stay resident in the 192MB L2).
// Fragment layouts per CDNA5 ISA 7.12.2:
//   A 16x4 : lanes 0-15 -> {K=k0,k0+1}, lanes 16-31 -> {K=k0+2,k0+3}, M=lane%16
//   B 4x16 : lanes 0-15 -> rows k0,k0+1 at N=lane, lanes 16-31 -> rows k0+2,k0+3
//   C/D    : VGPR i -> row i (lanes 0-15) / row i+8 (lanes 16-31), N=lane%16
// All WMMA issued under uniform control flow (EXEC all-ones).
// ---------------------------------------------------------------------------
__global__ __launch_bounds__(128) void sage_gemm(
    const float* __restrict__ A1, const float* __restrict__ cnt,
    const float* __restrict__ W1, int K1,
    const float* __restrict__ A2, const float* __restrict__ W2, int K2,
    const float* __restrict__ bias, float* __restrict__ out,
    int N, int do_relu) {
  __shared__ float sA1[16 * 128];
  __shared__ float sA2[16 * 128];
  __shared__ float sInv[16];

  const int tid = threadIdx.x;
  const long long m0 = (long long)blockIdx.x * 16;

  if (tid < 16) {
    float c = (cnt != nullptr) ? cnt[m0 + tid] : 1.0f;
    sInv[tid] = 1.0f / fmaxf(c, 1.0f);
  }
  __syncthreads();

  for (int i = tid; i < 16 * K1; i += 128) {
    int r = i / K1;
    int k = i - r * K1;
    sA1[r * K1 + k] = A1[(m0 + r) * K1 + k] * sInv[r];
  }
  if (A2 != nullptr) {
    for (int i = tid; i < 16 * K2; i += 128) {
      int r = i / K2;
      int k = i - r * K2;
      sA2[r * K2 + k] = A2[(m0 + r) * K2 + k];
    }
  }
  __syncthreads();

  const int wave = tid >> 5;
  const int lane = tid & 31;
  const int half = lane >> 4;   // 0: lanes 0-15, 1: lanes 16-31
  const int r    = lane & 15;
  const int ntiles = N >> 4;

  for (int t = wave; t < ntiles; t += 4) {
    const int nb = t << 4;
    v8f acc = {};
    for (int k0 = 0; k0 < K1; k0 += 4) {
      v2f a, b;
      a.x = sA1[r * K1 + k0 + 2 * half + 0];
      a.y = sA1[r * K1 + k0 + 2 * half + 1];
      b.x = W1[(long long)(k0 + 2 * half + 0) * N + nb + r];
      b.y = W1[(long long)(k0 + 2 * half + 1) * N + nb + r];
      acc = wmma_f32_k4(a, b, acc);
    }
    if (A2 != nullptr) {
      for (int k0 = 0; k0 < K2; k0 += 4) {
        v2f a, b;
        a.x = sA2[r * K2 + k0 + 2 * half + 0];
        a.y = sA2[r * K2 + k0 + 2 * half + 1];
        b.x = W2[(long long)(k0 + 2 * half + 0) * N + nb + r];
        b.y = W2[(long long)(k0 + 2 * half + 1) * N + nb + r];
        acc = wmma_f32_k4(a, b, acc);
      }
    }
    const float bv = bias[nb + r];
#pragma unroll
    for (int i = 0; i < 8; ++i) {
      float v = acc[i] + bv;
      if (do_relu) v = fmaxf(v, 0.0f);
      const int row = i + 8 * half;
      out[(m0 + row) * N + nb + r] = v;
    }
  }
}

// ---------------------------------------------------------------------------
extern "C" void kernel_launch(void* const* d_in, const int* in_sizes, int n_in,
                              void* d_out, int out_size, void* d_ws, size_t ws_size,
                              hipStream_t stream) {
  (void)n_in; (void)out_size; (void)ws_size;

  const float* x_txn   = (const float*)d_in[0];
  const float* x_cli   = (const float*)d_in[1];
  const int* ei_ct_src = (const int*)d_in[2];
  const int* ei_ct_dst = (const int*)d_in[3];
  const int* ei_tc_src = (const int*)d_in[4];
  const int* ei_tc_dst = (const int*)d_in[5];
  const float* W1_ct_l = (const float*)d_in[6];
  const float* b1_ct   = (const float*)d_in[7];
  const float* W1_ct_r = (const float*)d_in[8];
  const float* W1_tc_l = (const float*)d_in[9];
  const float* b1_tc   = (const float*)d_in[10];
  const float* W1_tc_r = (const float*)d_in[11];
  const float* W2_ct_l = (const float*)d_in[12];
  const float* b2_ct   = (const float*)d_in[13];
  const float* W2_ct_r = (const float*)d_in[14];
  const float* W2_tc_l = (const float*)d_in[15];
  const float* b2_tc   = (const float*)d_in[16];
  const float* W2_tc_r = (const float*)d_in[17];
  const float* Wd1     = (const float*)d_in[18];
  const float* bd1     = (const float*)d_in[19];
  const float* Wd2     = (const float*)d_in[20];
  const float* bd2     = (const float*)d_in[21];

  const int E = in_sizes[2];

  // Workspace layout: all scatter accumulators first (zeroed each call).
  float* ws = (float*)d_ws;
  long long off = 0;
  float* agg1_txn = ws + off; off += (long long)N_TXN_C * D_CLI_C;  // mean(client) per txn
  float* agg1_cli = ws + off; off += (long long)N_CLI_C * D_TXN_C;  // mean(txn) per client
  float* cnt_txn  = ws + off; off += N_TXN_C;
  float* cnt_cli  = ws + off; off += N_CLI_C;
  float* agg2_txn = ws + off; off += (long long)N_TXN_C * H_C;
  float* agg2_cli = ws + off; off += (long long)N_CLI_C * H_C;
  const long long n_zero = off;
  float* h_txn = ws + off; off += (long long)N_TXN_C * H_C;
  float* h_cli = ws + off; off += (long long)N_CLI_C * H_C;
  float* mid   = agg2_txn;  // alias: agg2_txn dead after z_txn GEMM

  // Output layout (concatenated flat in return order).
  float* recon = (float*)d_out;                               // [100000, 64]
  float* z_txn = recon + (long long)N_TXN_C * D_TXN_C;        // [100000, 128]
  float* z_cli = z_txn + (long long)N_TXN_C * H_C;            // [20000, 128]

  // 1) Zero scatter accumulators + degree counts.
  zero_f32<<<4096, 256, 0, stream>>>(ws, n_zero);

  const int sblocks = (E + 7) / 8;  // 8 edges (waves) per 256-thread block

  // 2) Layer-1 aggregation (+ degree counts, reused by layer 2).
  scatter_add<<<sblocks, 256, 0, stream>>>(x_cli, ei_ct_src, ei_ct_dst,
                                           agg1_txn, cnt_txn, E, D_CLI_C);
  scatter_add<<<sblocks, 256, 0, stream>>>(x_txn, ei_tc_src, ei_tc_dst,
                                           agg1_cli, cnt_cli, E, D_TXN_C);

  // 3) Layer-1 SAGE transforms (+ ReLU).
  sage_gemm<<<N_TXN_C / 16, 128, 0, stream>>>(agg1_txn, cnt_txn, W1_ct_l, D_CLI_C,
                                              x_txn, W1_ct_r, D_TXN_C,
                                              b1_ct, h_txn, H_C, 1);
  sage_gemm<<<N_CLI_C / 16, 128, 0, stream>>>(agg1_cli, cnt_cli, W1_tc_l, D_TXN_C,
                                              x_cli, W1_tc_r, D_CLI_C,
                                              b1_tc, h_cli, H_C, 1);

  // 4) Layer-2 aggregation (counts already known).
  scatter_add<<<sblocks, 256, 0, stream>>>(h_cli, ei_ct_src, ei_ct_dst,
                                           agg2_txn, nullptr, E, H_C);
  scatter_add<<<sblocks, 256, 0, stream>>>(h_txn, ei_tc_src, ei_tc_dst,
                                           agg2_cli, nullptr, E, H_C);

  // 5) Layer-2 SAGE transforms (no activation) -> outputs z_txn, z_cli.
  sage_gemm<<<N_TXN_C / 16, 128, 0, stream>>>(agg2_txn, cnt_txn, W2_ct_l, H_C,
                                              h_txn, W2_ct_r, H_C,
                                              b2_ct, z_txn, H_C, 0);
  sage_gemm<<<N_CLI_C / 16, 128, 0, stream>>>(agg2_cli, cnt_cli, W2_tc_l, H_C,
                                              h_cli, W2_tc_r, H_C,
                                              b2_tc, z_cli, H_C, 0);

  // 6) Decoder MLP on txn embeddings.
  sage_gemm<<<N_TXN_C / 16, 128, 0, stream>>>(z_txn, nullptr, Wd1, H_C,
                                              nullptr, nullptr, 0,
                                              bd1, mid, 64, 1);
  sage_gemm<<<N_TXN_C / 16, 128, 0, stream>>>(mid, nullptr, Wd2, 64,
                                              nullptr, nullptr, 0,
                                              bd2, recon, 64, 0);
}